// GATLayer_46858093199522
// MI455X (gfx1250) — compile-verified
//
#include <hip/hip_runtime.h>
#include <math.h>

#define N_NODES 100000
#define N_EDGES 800000
#define IN_DIM  128
#define HEADS   4
#define DHEAD   32
#define HD      128
#define EPS_LN  1e-5f
#define NEG_SLOPE 0.2f

typedef __attribute__((ext_vector_type(2))) float v2f;
typedef __attribute__((ext_vector_type(4))) float v4f;
typedef __attribute__((ext_vector_type(8))) float v8f;

// ---------------- K0: init out=0, esum=0, emax=-inf ----------------
__global__ void fill_kernel(float* __restrict__ out, float* __restrict__ esum,
                            int* __restrict__ emax) {
    int i = blockIdx.x * blockDim.x + threadIdx.x;
    if (i < N_NODES * HD) out[i] = 0.0f;
    if (i < N_NODES * HEADS) {
        esum[i] = 0.0f;
        emax[i] = (int)0xFF800000u;  // -inf bit pattern
    }
}

// ---------------- K1: WMMA f32 GEMM: feat_src = feat*Wsrc^T + bsrc, skip = feat*Wskip^T + bskip
__global__ __launch_bounds__(256) void gemm_wmma_kernel(
    const float* __restrict__ feat,
    const float* __restrict__ Wsrc,  const float* __restrict__ bsrc,
    const float* __restrict__ Wskip, const float* __restrict__ bskip,
    float* __restrict__ feat_src, float* __restrict__ skip) {
    int wave = (blockIdx.x * blockDim.x + threadIdx.x) >> 5;
    int lane = threadIdx.x & 31;
    if (wave >= N_NODES / 16) return;     // uniform per wave; EXEC stays all-ones
    const int node0 = wave * 16;
    const int half  = lane >> 4;          // 0 or 1
    const int l16   = lane & 15;

    // A fragment: row = l16, K pair base = 2*half (ISA 32-bit A 16x4 layout)
    const float* Arow = feat + (size_t)(node0 + l16) * IN_DIM + 2 * half;

    for (int p = 0; p < 2; ++p) {
        const float* W    = p ? Wskip : Wsrc;
        const float* bias = p ? bskip : bsrc;
        float*       out  = p ? skip  : feat_src;
        for (int ct = 0; ct < 8; ++ct) {          // 8 column tiles of 16
            const int n = ct * 16 + l16;          // output channel for this lane
            v8f acc;
            const float bv = bias[n];
            #pragma unroll
            for (int v = 0; v < 8; ++v) acc[v] = bv;   // bias preloaded into C
            // B fragment: B[k][n] = W[n][k]; per lane K pair base = 2*half
            const float* Bcol = W + (size_t)n * IN_DIM + 2 * half;
            for (int kt = 0; kt < 32; ++kt) {     // K = 128 = 32 * 4
                v2f a = *(const v2f*)(Arow + kt * 4);
                v2f b = *(const v2f*)(Bcol + kt * 4);
                acc = __builtin_amdgcn_wmma_f32_16x16x4_f32(
                    false, a, false, b, (short)0, acc, false, false);
            }
            // C/D layout: VGPR v holds (m = v + 8*half, n = ct*16 + l16)
            float* obase = out + (size_t)node0 * HD + ct * 16 + l16;
            #pragma unroll
            for (int v = 0; v < 8; ++v)
                obase[(size_t)(v + 8 * half) * HD] = acc[v];
        }
    }
}

// ---------------- K2: el/er per (node, head) ----------------
__global__ void elr_kernel(const float* __restrict__ feat_src,
                           const float* __restrict__ attn_l,
                           const float* __restrict__ attn_r,
                           float* __restrict__ el, float* __restrict__ er) {
    int i = blockIdx.x * blockDim.x + threadIdx.x;   // i = n*H + h
    if (i >= N_NODES * HEADS) return;
    const int h = i & 3;
    const float* f  = feat_src + (size_t)(i >> 2) * HD + h * DHEAD;
    const float* al = attn_l + h * DHEAD;
    const float* ar = attn_r + h * DHEAD;
    float sl = 0.0f, sr = 0.0f;
    #pragma unroll
    for (int d = 0; d < DHEAD; ++d) {
        float v = f[d];
        sl += v * al[d];
        sr += v * ar[d];
    }
    el[i] = sl;
    er[i] = sr;
}

// ---------------- K3: edge logit + leaky relu + atomic float max ----------------
__global__ void edge_logit_kernel(const int* __restrict__ src, const int* __restrict__ dst,
                                  const float* __restrict__ el, const float* __restrict__ er,
                                  float* __restrict__ ebuf, int* __restrict__ emax) {
    int e = blockIdx.x * blockDim.x + threadIdx.x;
    if (e >= N_EDGES) return;
    const int s = src[e], d = dst[e];
    const v4f els = *(const v4f*)(el + (size_t)s * 4);
    const v4f erd = *(const v4f*)(er + (size_t)d * 4);
    #pragma unroll
    for (int h = 0; h < 4; ++h) {
        float v = els[h] + erd[h];
        v = (v >= 0.0f) ? v : NEG_SLOPE * v;
        ebuf[(size_t)e * 4 + h] = v;
        int* p = emax + (size_t)d * 4 + h;
        if (v >= 0.0f) atomicMax(p, __float_as_int(v));
        else           atomicMin((unsigned int*)p, __float_as_uint(v));
    }
}

// ---------------- K4: exp(e - emax[dst]) + segment sum ----------------
__global__ void edge_exp_kernel(const int* __restrict__ dst,
                                const float* __restrict__ emax,
                                float* __restrict__ ebuf, float* __restrict__ esum) {
    int e = blockIdx.x * blockDim.x + threadIdx.x;
    if (e >= N_EDGES) return;
    const int d = dst[e];
    #pragma unroll
    for (int h = 0; h < 4; ++h) {
        float m  = emax[(size_t)d * 4 + h];
        float ex = __expf(ebuf[(size_t)e * 4 + h] - m);
        ebuf[(size_t)e * 4 + h] = ex;
        atomicAdd(esum + (size_t)d * 4 + h, ex);
    }
}

// ---------------- K5: aggregation, one wave per edge ----------------
__global__ __launch_bounds__(256) void aggregate_kernel(
    const int* __restrict__ src, const int* __restrict__ dst,
    const float* __restrict__ ebuf, const float* __restrict__ feat_src,
    float* __restrict__ out) {
    int wave = (blockIdx.x * blockDim.x + threadIdx.x) >> 5;
    int lane = threadIdx.x & 31;
    if (wave >= N_EDGES) return;
    const int s = src[wave], d = dst[wave];
    const int h = lane >> 3;                   // channels lane*4..lane*4+3 -> head
    const float a = ebuf[(size_t)wave * 4 + h];
    const v4f f = *(const v4f*)(feat_src + (size_t)s * HD + lane * 4);
    float* o = out + (size_t)d * HD + lane * 4;
    atomicAdd(o + 0, a * f.x);
    atomicAdd(o + 1, a * f.y);
    atomicAdd(o + 2, a * f.z);
    atomicAdd(o + 3, a * f.w);
}

// ---------------- K6: fused normalize + gate + LayerNorm + PReLU, wave/node ----------------
__global__ __launch_bounds__(256) void final_kernel(
    const float* __restrict__ skip, const float* __restrict__ esum,
    const float* __restrict__ Wg, const float* __restrict__ bg,
    const float* __restrict__ ln_g, const float* __restrict__ ln_b,
    const float* __restrict__ prelu_a, float* __restrict__ out) {
    int wave = (blockIdx.x * blockDim.x + threadIdx.x) >> 5;
    int lane = threadIdx.x & 31;
    if (wave >= N_NODES) return;
    const int c = lane * 4;
    const int h = lane >> 3;
    const float es  = esum[(size_t)wave * 4 + h];
    const float inv = (es > 0.0f) ? 1.0f / es : 0.0f;   // empty segments -> 0
    v4f r  = *(const v4f*)(out  + (size_t)wave * HD + c);
    r = r * inv;
    const v4f sk = *(const v4f*)(skip + (size_t)wave * HD + c);
    const v4f dm = r - sk;
    const v4f w0 = *(const v4f*)(Wg + c);
    const v4f w1 = *(const v4f*)(Wg + 128 + c);
    const v4f w2 = *(const v4f*)(Wg + 256 + c);
    float part = r.x * w0.x + r.y * w0.y + r.z * w0.z + r.w * w0.w
               + sk.x * w1.x + sk.y * w1.y + sk.z * w1.z + sk.w * w1.w
               + dm.x * w2.x + dm.y * w2.y + dm.z * w2.z + dm.w * w2.w;
    #pragma unroll
    for (int off = 16; off > 0; off >>= 1) part += __shfl_xor(part, off, 32);
    const float g = 1.0f / (1.0f + __expf(-(part + bg[0])));
    v4f z = g * r + (1.0f - g) * sk;
    float s  = z.x + z.y + z.z + z.w;
    float sq = z.x * z.x + z.y * z.y + z.z * z.z + z.w * z.w;
    #pragma unroll
    for (int off = 16; off > 0; off >>= 1) {
        s  += __shfl_xor(s,  off, 32);
        sq += __shfl_xor(sq, off, 32);
    }
    const float mu   = s * (1.0f / 128.0f);
    const float var  = sq * (1.0f / 128.0f) - mu * mu;
    const float rstd = rsqrtf(var + EPS_LN);
    const float alpha = prelu_a[0];
    const v4f lg = *(const v4f*)(ln_g + c);
    const v4f lb = *(const v4f*)(ln_b + c);
    v4f y;
    #pragma unroll
    for (int i = 0; i < 4; ++i) {
        float yy = (z[i] - mu) * rstd * lg[i] + lb[i];
        y[i] = (yy >= 0.0f) ? yy : alpha * yy;
    }
    *(v4f*)(out + (size_t)wave * HD + c) = y;
}

extern "C" void kernel_launch(void* const* d_in, const int* in_sizes, int n_in,
                              void* d_out, int out_size, void* d_ws, size_t ws_size,
                              hipStream_t stream) {
    const float* feat    = (const float*)d_in[0];
    const int*   src     = (const int*)  d_in[1];
    const int*   dst     = (const int*)  d_in[2];
    const float* W_src   = (const float*)d_in[3];
    const float* b_src   = (const float*)d_in[4];
    const float* attn_l  = (const float*)d_in[5];
    const float* attn_r  = (const float*)d_in[6];
    const float* W_skip  = (const float*)d_in[7];
    const float* b_skip  = (const float*)d_in[8];
    const float* W_gate  = (const float*)d_in[9];
    const float* b_gate  = (const float*)d_in[10];
    const float* ln_g    = (const float*)d_in[11];
    const float* ln_b    = (const float*)d_in[12];
    const float* prelu_a = (const float*)d_in[13];
    float* out = (float*)d_out;

    // workspace layout (floats)
    float* ws       = (float*)d_ws;
    float* feat_src = ws;                                        // N*128
    float* skip     = feat_src + (size_t)N_NODES * HD;           // N*128
    float* el       = skip     + (size_t)N_NODES * HD;           // N*4
    float* er       = el       + (size_t)N_NODES * HEADS;        // N*4
    float* emax     = er       + (size_t)N_NODES * HEADS;        // N*4 (int bits)
    float* esum     = emax     + (size_t)N_NODES * HEADS;        // N*4
    float* ebuf     = esum     + (size_t)N_NODES * HEADS;        // E*4

    // K0: init
    {
        int n = N_NODES * HD;
        fill_kernel<<<(n + 255) / 256, 256, 0, stream>>>(out, esum, (int*)emax);
    }
    // K1: WMMA GEMMs
    {
        int waves  = N_NODES / 16;                   // 6250
        int blocks = (waves * 32 + 255) / 256;
        gemm_wmma_kernel<<<blocks, 256, 0, stream>>>(feat, W_src, b_src, W_skip, b_skip,
                                                     feat_src, skip);
    }
    // K2: el/er
    {
        int n = N_NODES * HEADS;
        elr_kernel<<<(n + 255) / 256, 256, 0, stream>>>(feat_src, attn_l, attn_r, el, er);
    }
    // K3: edge logits + max
    edge_logit_kernel<<<(N_EDGES + 255) / 256, 256, 0, stream>>>(src, dst, el, er, ebuf,
                                                                 (int*)emax);
    // K4: exp + sum
    edge_exp_kernel<<<(N_EDGES + 255) / 256, 256, 0, stream>>>(dst, emax, ebuf, esum);
    // K5: aggregation (wave per edge)
    {
        int blocks = (N_EDGES * 32 + 255) / 256;
        aggregate_kernel<<<blocks, 256, 0, stream>>>(src, dst, ebuf, feat_src, out);
    }
    // K6: fused epilogue (wave per node)
    {
        int blocks = (N_NODES * 32 + 255) / 256;
        final_kernel<<<blocks, 256, 0, stream>>>(skip, esum, W_gate, b_gate, ln_g, ln_b,
                                                 prelu_a, out);
    }
}